// MarginalGaussianization_82128364634361
// MI455X (gfx1250) — compile-verified
//
#include <hip/hip_runtime.h>
#include <hip/hip_bf16.h>
#include <stdint.h>

// Problem constants (from the reference setup_inputs()).
#define B_TOTAL 262144
#define D_DIM   64
#define NB      1000
#define NB4     (NB / 4)          // 250 float4 per table row
#define CD      4                 // dims per LDS chunk
#define NCHUNK  (D_DIM / CD)      // 16 chunks
#define ROWP    1024              // padded LDS row length (floats)
#define BLOCK   256

// ---------------- CDNA5 async global->LDS copy (ASYNCcnt path) -------------
// GV-mode form: vdst = VGPR holding LDS byte address, vaddr = 64-bit global
// address VGPR pair, saddr = off.  Tracked by ASYNCcnt.
__device__ __forceinline__ void async_ld_b128(unsigned lds_off, const float* gaddr) {
    asm volatile("global_load_async_to_lds_b128 %0, %1, off"
                 :: "v"(lds_off), "v"(gaddr)
                 : "memory");
}

__device__ __forceinline__ void wait_async0() {
    asm volatile("s_wait_asynccnt 0" ::: "memory");
}

// ---------------- single-precision erfinv (Giles 2010) ----------------------
__device__ __forceinline__ float erfinv_f(float x) {
    float w = -__logf((1.0f - x) * (1.0f + x));
    float p;
    if (w < 5.0f) {
        w = w - 2.5f;
        p =  2.81022636e-08f;
        p =  3.43273939e-07f + p * w;
        p = -3.52338770e-06f + p * w;
        p = -4.39150654e-06f + p * w;
        p =  2.18580870e-04f + p * w;
        p = -1.25372503e-03f + p * w;
        p = -4.17768164e-03f + p * w;
        p =  2.46640727e-01f + p * w;
        p =  1.50140941e+00f + p * w;
    } else {
        w = __fsqrt_rn(w) - 3.0f;
        p = -2.00214257e-04f;
        p =  1.00950558e-04f + p * w;
        p =  1.34934322e-03f + p * w;
        p = -3.67342844e-03f + p * w;
        p =  5.73950773e-03f + p * w;
        p = -7.62246130e-03f + p * w;
        p =  9.43887047e-03f + p * w;
        p =  1.00167406e+00f + p * w;
        p =  2.83297682e+00f + p * w;
    }
    return p * x;
}

__global__ __launch_bounds__(BLOCK)
void marginal_gaussianization_kernel(const float* __restrict__ x,
                                     const float* __restrict__ xval,
                                     const float* __restrict__ cdfval,
                                     float* __restrict__ z_out,
                                     float* __restrict__ ld_out) {
    // [buffer][table (0=x_values,1=cdf)][local dim][padded row] : 64 KB total
    __shared__ float smem[2][2][CD][ROWP];

    const int  tid = threadIdx.x;
    const long b   = (long)blockIdx.x * BLOCK + tid;

    const float* xrow = x     + b * D_DIM;
    float*       zrow = z_out + b * D_DIM;

    // Speculative prefetch of this sample's x row (global_prefetch_b8).
    __builtin_prefetch(xrow, 0, 0);

    // Issue async copies of one chunk's tables into LDS buffer `buf`.
    // Thread = one float4 column (250 active lanes), nested loops over
    // table/row: no integer division, adjacent lanes -> adjacent 16B.
    auto issue_chunk = [&](int c, int buf) {
        if (tid < NB4) {
            const size_t col = (size_t)tid * 4;
            #pragma unroll
            for (int t = 0; t < 2; ++t) {
                const float* tab = (t == 0) ? xval : cdfval;
                #pragma unroll
                for (int r = 0; r < CD; ++r) {
                    const float* src = tab + (size_t)(c * CD + r) * NB + col;
                    const unsigned dst =
                        (unsigned)(uintptr_t)&smem[buf][t][r][col];
                    async_ld_b128(dst, src);
                }
            }
        }
    };

    // Prime the pipeline with chunk 0.
    issue_chunk(0, 0);

    const float inv_sqrt_2pi = 0.3989422804014327f;
    const float sqrt2        = 1.4142135623730951f;
    float logdet = 0.0f;

    for (int c = 0; c < NCHUNK; ++c) {
        // Wave waits for its own async copies; barrier publishes all waves'.
        wait_async0();
        __syncthreads();

        // Overlap: issue next chunk's copies into the other buffer while we
        // compute on the current one.
        if (c + 1 < NCHUNK) issue_chunk(c + 1, (c + 1) & 1);

        const int buf = c & 1;

        const float4 xin = *reinterpret_cast<const float4*>(xrow + c * CD);
        const float* xin_s = reinterpret_cast<const float*>(&xin);
        float4 zvec;
        float* z_s = reinterpret_cast<float*>(&zvec);

        #pragma unroll
        for (int j = 0; j < CD; ++j) {
            const float* xs = &smem[buf][0][j][0];
            const float* cs = &smem[buf][1][j][0];
            const float  xd = xin_s[j];

            // Branchless rank search: pos = #{i : xs[i] < xd}
            // == jnp.searchsorted(xs, xd, side='left').
            // Non-power-of-2 prologue: pos in {0, 488}; steps 256..1 then run
            // unguarded (max probe index 998).  rank==1000 lands at 999,
            // identical after clip(idx, 1, 999).
            int pos = (xs[487] < xd) ? (NB - 512) : 0;
            #pragma unroll
            for (int s = 256; s >= 1; s >>= 1) {
                if (xs[pos + s - 1] < xd) pos += s;
            }
            const int idx = min(max(pos, 1), NB - 1);

            const float xl = xs[idx - 1];
            const float xr = xs[idx];
            const float cl = cs[idx - 1];
            const float cr = cs[idx];

            const float denom = xr - xl + 1e-12f;
            const float rden  = __frcp_rn(denom);
            float u = cl + (cr - cl) * (xd - xl) * rden;
            const float p_hat = fmaxf((cr - cl) * rden, 1e-12f);
            u = fminf(fmaxf(u, 1e-6f), 1.0f - 1e-6f);

            const float e  = fminf(fmaxf(2.0f * u - 1.0f, -0.99999f), 0.99999f);
            const float zv = fminf(fmaxf(sqrt2 * erfinv_f(e), -10.0f), 10.0f);
            const float phi = inv_sqrt_2pi * __expf(-0.5f * zv * zv);

            logdet += __logf(p_hat) - __logf(phi + 1e-12f);
            z_s[j] = zv;
        }

        *reinterpret_cast<float4*>(zrow + c * CD) = zvec;

        // All threads must finish reading this buffer before it is refilled
        // (the next iteration refills it with chunk c + 2).
        __syncthreads();
    }

    ld_out[b] = logdet;
}

extern "C" void kernel_launch(void* const* d_in, const int* in_sizes, int n_in,
                              void* d_out, int out_size, void* d_ws, size_t ws_size,
                              hipStream_t stream) {
    const float* x      = (const float*)d_in[0];   // [B, 64]
    const float* xval   = (const float*)d_in[1];   // [64, 1000]
    const float* cdfval = (const float*)d_in[2];   // [64, 1000]

    float* z_out  = (float*)d_out;                         // [B, 64]
    float* ld_out = z_out + (size_t)B_TOTAL * D_DIM;       // [B]

    dim3 grid(B_TOTAL / BLOCK);
    marginal_gaussianization_kernel<<<grid, BLOCK, 0, stream>>>(
        x, xval, cdfval, z_out, ld_out);
}